// RatledgeLoss_7310034338547
// MI455X (gfx1250) — compile-verified
//
#include <hip/hip_runtime.h>

typedef float v2f __attribute__((ext_vector_type(2)));
typedef float v4f __attribute__((ext_vector_type(4)));
typedef float v8f __attribute__((ext_vector_type(8)));

// ---------------------------------------------------------------------------
// wave32 butterfly sum (exact f32, 5 steps)
// ---------------------------------------------------------------------------
__device__ __forceinline__ float wave_sum(float v) {
#pragma unroll
  for (int off = 16; off > 0; off >>= 1) v += __shfl_xor(v, off, 32);
  return v;
}

// ---------------------------------------------------------------------------
// Kernel 1: streaming pass. 256 threads/block = 8 wave32.
//   per-lane accumulators -> wave butterfly -> LDS rows (8 waves x 16 cols)
//   -> cross-wave reduce with 2x V_WMMA_F32_16X16X4_F32 (A = ones):
//      D[m][n] = sum_k B[k][n]  (exact f32 column sums, rows 0..3 then 4..7)
//   block totals land in D row 0 (lanes 0..15 of C[0]) -> ws[block*16 + j].
// ---------------------------------------------------------------------------
__global__ __launch_bounds__(256) void ratledge_partial(
    const float* __restrict__ yp, const float* __restrict__ yt,
    const float* __restrict__ q, float* __restrict__ ws,
    int nvec, int n) {
  const float q0 = q[0], q1 = q[1], q2 = q[2], q3 = q[3], q4 = q[4], q5 = q[5];

  float sumsq = 0.f;
  float bs0 = 0.f, bs1 = 0.f, bs2 = 0.f, bs3 = 0.f, bs4 = 0.f;
  float bc0 = 0.f, bc1 = 0.f, bc2 = 0.f, bc3 = 0.f, bc4 = 0.f;

  auto proc = [&](float p, float t) {
    float d = p - t;
    sumsq = __builtin_fmaf(d, d, sumsq);
    // searchsorted(quants, t, side='right') - 1, closed last bin:
    // idx = #{j in 1..4 : t >= q_j}; valid iff q0 <= t <= q5.
    bool valid = (t >= q0) && (t <= q5);
    int idx = (int)(t >= q1) + (int)(t >= q2) + (int)(t >= q3) + (int)(t >= q4);
    bool m0 = valid && (idx == 0); bs0 += m0 ? d : 0.f; bc0 += m0 ? 1.f : 0.f;
    bool m1 = valid && (idx == 1); bs1 += m1 ? d : 0.f; bc1 += m1 ? 1.f : 0.f;
    bool m2 = valid && (idx == 2); bs2 += m2 ? d : 0.f; bc2 += m2 ? 1.f : 0.f;
    bool m3 = valid && (idx == 3); bs3 += m3 ? d : 0.f; bc3 += m3 ? 1.f : 0.f;
    bool m4 = valid && (idx == 4); bs4 += m4 ? d : 0.f; bc4 += m4 ? 1.f : 0.f;
  };

  const int gtid   = blockIdx.x * blockDim.x + threadIdx.x;
  const int stride = gridDim.x * blockDim.x;

  // main body: b128 non-temporal streaming loads (no reuse, bypass caches)
  const v4f* yp4 = (const v4f*)yp;
  const v4f* yt4 = (const v4f*)yt;
  for (int i = gtid; i < nvec; i += stride) {
    v4f p = __builtin_nontemporal_load(&yp4[i]);
    v4f t = __builtin_nontemporal_load(&yt4[i]);
#pragma unroll
    for (int e = 0; e < 4; ++e) proc(p[e], t[e]);
  }
  // scalar tail
  for (int i = (nvec << 2) + gtid; i < n; i += stride)
    proc(__builtin_nontemporal_load(&yp[i]), __builtin_nontemporal_load(&yt[i]));

  // ---- wave reduction (exact) ----
  sumsq = wave_sum(sumsq);
  bs0 = wave_sum(bs0); bs1 = wave_sum(bs1); bs2 = wave_sum(bs2);
  bs3 = wave_sum(bs3); bs4 = wave_sum(bs4);
  bc0 = wave_sum(bc0); bc1 = wave_sum(bc1); bc2 = wave_sum(bc2);
  bc3 = wave_sum(bc3); bc4 = wave_sum(bc4);

  __shared__ float red[8][16];
  const int wave = threadIdx.x >> 5;
  const int lane = threadIdx.x & 31;
  if (lane == 0) {
    red[wave][0] = sumsq;
    red[wave][1] = bs0; red[wave][2] = bs1; red[wave][3] = bs2;
    red[wave][4] = bs3; red[wave][5] = bs4;
    red[wave][6] = bc0; red[wave][7] = bc1; red[wave][8] = bc2;
    red[wave][9] = bc3; red[wave][10] = bc4;
#pragma unroll
    for (int j = 11; j < 16; ++j) red[wave][j] = 0.f;
  }
  __syncthreads();

  // ---- cross-wave reduction on the matrix path (wave 0 only, EXEC all-1) ----
  if (wave == 0) {
    const int col = lane & 15;
    const int h2  = (lane >> 4) << 1;  // 0 for lanes 0-15, 2 for lanes 16-31
    v2f A;  A[0] = 1.0f; A[1] = 1.0f;                       // 16x4 of ones
    v2f B0; B0[0] = red[h2 + 0][col]; B0[1] = red[h2 + 1][col];  // rows 0,1 / 2,3
    v2f B1; B1[0] = red[h2 + 4][col]; B1[1] = red[h2 + 5][col];  // rows 4,5 / 6,7
    v8f C = {};
    C = __builtin_amdgcn_wmma_f32_16x16x4_f32(false, A, false, B0,
                                              (short)0, C, false, false);
    C = __builtin_amdgcn_wmma_f32_16x16x4_f32(false, A, false, B1,
                                              (short)0, C, false, false);
    // D VGPR0, lanes 0..15 = D[0][n] = block total for accumulator n
    if (lane < 16) ws[blockIdx.x * 16 + lane] = C[0];
  }
}

// ---------------------------------------------------------------------------
// Kernel 2: fold block rows (deterministic order) + scalar epilogue
// ---------------------------------------------------------------------------
__global__ __launch_bounds__(64) void ratledge_final(
    const float* __restrict__ ws, float* __restrict__ out, int nb, float fn) {
  __shared__ float tot[16];
  const int t = threadIdx.x;
  if (t < 16) {
    float s = 0.f;
    for (int b = 0; b < nb; ++b) s += ws[b * 16 + t];
    tot[t] = s;
  }
  __syncthreads();
  if (t == 0) {
    float mse = tot[0] / fn;
    float mx = 0.f;
#pragma unroll
    for (int j = 0; j < 5; ++j) {
      float c    = tot[6 + j];
      float bias = tot[1 + j] / fmaxf(c, 1.f);
      float b2   = (c > 0.f) ? bias * bias : 0.f;
      mx = fmaxf(mx, b2);
    }
    out[0] = mse + 5.0f * mx;  // LAMBDA_B = 5.0
  }
}

// ---------------------------------------------------------------------------
extern "C" void kernel_launch(void* const* d_in, const int* in_sizes, int n_in,
                              void* d_out, int out_size, void* d_ws, size_t ws_size,
                              hipStream_t stream) {
  const float* yp = (const float*)d_in[0];
  const float* yt = (const float*)d_in[1];
  const float* q  = (const float*)d_in[2];
  float* out = (float*)d_out;
  float* ws  = (float*)d_ws;

  const int n    = in_sizes[0];
  const int nvec = n >> 2;

  int nb = (int)(ws_size / (16 * sizeof(float)));
  if (nb > 1024) nb = 1024;   // 1024 blocks x 256 thr = 256K threads, 16 iters each
  if (nb < 1)   nb = 1;

  ratledge_partial<<<nb, 256, 0, stream>>>(yp, yt, q, ws, nvec, n);
  ratledge_final<<<1, 64, 0, stream>>>(ws, out, nb, (float)n);
}